// GATModule_7121055776909
// MI455X (gfx1250) — compile-verified
//
#include <hip/hip_runtime.h>
#include <hip/hip_bf16.h>

typedef __attribute__((ext_vector_type(2))) float v2f;
typedef __attribute__((ext_vector_type(8))) float v8f;

#define KDIM 256   // inner dim is 256 for both layers (IN=256, H*HID=256)
#define CDIM 256   // output feature dim for both layers (H*HID = H*OUT = 256)

// ---------------------------------------------------------------------------
// monotonic float<->uint encoding so we can use integer atomicMax for the
// segment-max (works for negative values, deterministic).
__device__ __forceinline__ unsigned enc_f32(float f) {
  unsigned u = __float_as_uint(f);
  return (u & 0x80000000u) ? ~u : (u | 0x80000000u);
}
__device__ __forceinline__ float dec_f32(unsigned u) {
  unsigned v = (u & 0x80000000u) ? (u & 0x7FFFFFFFu) : ~u;
  return __uint_as_float(v);
}

// Async global->LDS copy of 16 bytes (CDNA5 GLOBAL_LOAD_ASYNC_TO_LDS_B128,
// GV addressing mode, tracked by ASYNCcnt). lds_addr is the low-32-bit LDS
// byte address (generic-pointer truncation per the aperture rules).
__device__ __forceinline__ void async_copy_b128(unsigned lds_addr, const void* gptr) {
  asm volatile("global_load_async_to_lds_b128 %0, %1, off"
               :: "v"(lds_addr), "v"(gptr)
               : "memory");
}
__device__ __forceinline__ void wait_async_lds() {
  asm volatile("s_wait_asynccnt 0" ::: "memory");
}

// ---------------------------------------------------------------------------
// FP32 WMMA GEMM: C[M,256] = A[M,256] @ W[256,256], M multiple of 16.
// Block = 64 threads (2 waves). Each wave owns one 16x16 output tile.
// A tile (16x256, 16KB) and W tile (256x32, 32KB) staged in LDS via the
// async copy engine (no VGPR round-trip), then consumed by
// v_wmma_f32_16x16x4_f32 over a K-loop of 64 steps.
__global__ __launch_bounds__(64) void gemm_f32_wmma(
    const float* __restrict__ A, const float* __restrict__ W,
    float* __restrict__ C, int M) {
  __shared__ float ldsA[16 * KDIM];   // 16 KB
  __shared__ float ldsW[KDIM * 32];   // 32 KB
  const int tid   = threadIdx.x;
  const int lane  = tid & 31;
  const int wave  = tid >> 5;          // 0..1
  const int rowTile = blockIdx.x;      // 16 rows
  const int colBase = blockIdx.y * 32; // this block's 32 columns

  // A tile: 16x256 floats = 1024 x b128, 16 per thread, contiguous
  const char* Ag = (const char*)(A + (size_t)rowTile * 16 * KDIM);
  const unsigned ldsA_base = (unsigned)(size_t)&ldsA[0];
  for (int i = tid; i < 1024; i += 64)
    async_copy_b128(ldsA_base + (unsigned)i * 16u, Ag + (size_t)i * 16);

  // W tile: 256 rows x 32 cols = 2048 x b128 (8 b128 per row)
  const char* Wg = (const char*)(W + colBase);
  const unsigned ldsW_base = (unsigned)(size_t)&ldsW[0];
  for (int i = tid; i < 2048; i += 64) {
    int r = i >> 3, c4 = i & 7;
    async_copy_b128(ldsW_base + (unsigned)i * 16u,
                    Wg + (size_t)r * (CDIM * 4) + (size_t)c4 * 16);
  }
  wait_async_lds();     // our own async copies landed in LDS
  __syncthreads();      // all waves' copies visible

  const int halfsel = lane >> 4;   // 0: K=k,k+1 rows M | 1: K=k+2,k+3 rows M+8 (on store)
  const int l16     = lane & 15;
  const int wcol    = wave * 16 + l16;   // column within LDS W tile

  v8f acc = {};
#pragma unroll 8
  for (int k = 0; k < KDIM; k += 4) {
    const int ka = k + 2 * halfsel;
    v2f a, b;
    a.x = ldsA[l16 * KDIM + ka];
    a.y = ldsA[l16 * KDIM + ka + 1];
    b.x = ldsW[ka * 32 + wcol];
    b.y = ldsW[(ka + 1) * 32 + wcol];
    // D = A(16x4) x B(4x16) + C, full fp32  -> v_wmma_f32_16x16x4_f32
    acc = __builtin_amdgcn_wmma_f32_16x16x4_f32(
        false, a, false, b, (short)0, acc, false, false);
  }

  // C/D layout: VGPR i -> row (i + 8*halfsel), lane%16 -> col
  float* Cp = C + ((size_t)rowTile * 16 + 8 * halfsel) * CDIM + colBase + wcol;
#pragma unroll
  for (int i = 0; i < 8; ++i) Cp[(size_t)i * CDIM] = acc[i];
}

// ---------------------------------------------------------------------------
// a_src[n,h] = <xl[n,h,:], att_src[h,:]>, a_dst likewise. One thread per (n,h).
__global__ void attn_coef(const float* __restrict__ XL,
                          const float* __restrict__ as_, const float* __restrict__ ad_,
                          float* __restrict__ Asrc, float* __restrict__ Adst, int n) {
  int i = blockIdx.x * blockDim.x + threadIdx.x;
  if (i >= 2 * n) return;
  int head = i & 1, node = i >> 1;
  const float4* xr = (const float4*)(XL + (size_t)node * CDIM + head * 128);
  const float4* av = (const float4*)(as_ + head * 128);
  const float4* dv = (const float4*)(ad_ + head * 128);
  float s = 0.f, d = 0.f;
#pragma unroll 8
  for (int j = 0; j < 32; ++j) {
    float4 x = xr[j], a = av[j], b = dv[j];
    s += x.x * a.x + x.y * a.y + x.z * a.z + x.w * a.w;
    d += x.x * b.x + x.y * b.y + x.z * b.z + x.w * b.w;
  }
  Asrc[i] = s;
  Adst[i] = d;
}

// ---------------------------------------------------------------------------
__global__ void fill_u32(unsigned* __restrict__ p, unsigned v, size_t n) {
  size_t i = (size_t)blockIdx.x * blockDim.x + threadIdx.x;
  if (i < n) p[i] = v;
}
__global__ void fill_f32(float* __restrict__ p, float v, size_t n) {
  size_t i = (size_t)blockIdx.x * blockDim.x + threadIdx.x;
  if (i < n) p[i] = v;
}

__device__ __forceinline__ void edge_ids(int e, int E,
                                         const long long* __restrict__ es,
                                         const long long* __restrict__ ed,
                                         int& s, int& d) {
  if (e < E) { s = (int)es[e]; d = (int)ed[e]; }
  else       { s = e - E; d = e - E; }          // appended self loops
}

// pass 1: segment max of leaky_relu(a_src[s]+a_dst[d]) over dst
__global__ void edge_max_k(const long long* __restrict__ es, const long long* __restrict__ ed,
                           const float* __restrict__ Asrc, const float* __restrict__ Adst,
                           unsigned* __restrict__ emax, int E, int Nn) {
  int e = blockIdx.x * blockDim.x + threadIdx.x;
  if (e >= E + Nn) return;
  int s, d; edge_ids(e, E, es, ed, s, d);
#pragma unroll
  for (int h = 0; h < 2; ++h) {
    float v = Asrc[2 * s + h] + Adst[2 * d + h];
    v = (v > 0.f) ? v : 0.2f * v;
    atomicMax(&emax[2 * d + h], enc_f32(v));
  }
}

// pass 2: ee = exp(e - max[dst]); denom[dst] += ee
__global__ void edge_exp_k(const long long* __restrict__ es, const long long* __restrict__ ed,
                           const float* __restrict__ Asrc, const float* __restrict__ Adst,
                           const unsigned* __restrict__ emax,
                           float* __restrict__ ee, float* __restrict__ denom, int E, int Nn) {
  int e = blockIdx.x * blockDim.x + threadIdx.x;
  if (e >= E + Nn) return;
  int s, d; edge_ids(e, E, es, ed, s, d);
#pragma unroll
  for (int h = 0; h < 2; ++h) {
    float v = Asrc[2 * s + h] + Adst[2 * d + h];
    v = (v > 0.f) ? v : 0.2f * v;
    float x = __expf(v - dec_f32(emax[2 * d + h]));
    ee[2 * e + h] = x;
    atomicAdd(&denom[2 * d + h], x);
  }
}

// pass 3: ACC[dst] += XL[src] * alpha ; one wave32 per edge, 8 channels/lane
__global__ __launch_bounds__(256) void edge_scatter_k(
    const long long* __restrict__ es, const long long* __restrict__ ed,
    const float* __restrict__ XL, const float* __restrict__ ee,
    const float* __restrict__ denom, float* __restrict__ ACC, int E, int Nn) {
  int wid  = (int)((blockIdx.x * blockDim.x + threadIdx.x) >> 5);
  int lane = threadIdx.x & 31;
  if (wid >= E + Nn) return;
  int s, d; edge_ids(wid, E, es, ed, s, d);
  float a0 = ee[2 * wid + 0] / (denom[2 * d + 0] + 1e-16f);
  float a1 = ee[2 * wid + 1] / (denom[2 * d + 1] + 1e-16f);
  const float* xs = XL + (size_t)s * CDIM;
  float* od = ACC + (size_t)d * CDIM;
#pragma unroll
  for (int j = 0; j < 8; ++j) {
    int c = lane + 32 * j;
    float al = (c < 128) ? a0 : a1;
    atomicAdd(&od[c], xs[c] * al);
  }
}

// h = sigmoid(acc + bias[c]) in place
__global__ void bias_sigmoid_k(float* __restrict__ H, const float* __restrict__ b, size_t n) {
  size_t i = (size_t)blockIdx.x * blockDim.x + threadIdx.x;
  if (i >= n) return;
  float v = H[i] + b[i & (CDIM - 1)];
  H[i] = 1.f / (1.f + __expf(-v));
}

// out[a, 0:256] = H0[agent[a]], out[a, 256:512] = H1[agent[a]]
__global__ void gather_out_k(const long long* __restrict__ agent,
                             const float* __restrict__ H0, const float* __restrict__ H1,
                             float* __restrict__ out, int na) {
  int i = blockIdx.x * blockDim.x + threadIdx.x;
  if (i >= na * 512) return;
  int a = i >> 9, c = i & 511;
  long long node = agent[a];
  out[i] = (c < 256) ? H0[(size_t)node * CDIM + c] : H1[(size_t)node * CDIM + (c - 256)];
}

// ---------------------------------------------------------------------------
extern "C" void kernel_launch(void* const* d_in, const int* in_sizes, int n_in,
                              void* d_out, int out_size, void* d_ws, size_t ws_size,
                              hipStream_t stream) {
  const float*     x     = (const float*)d_in[0];
  const long long* ei    = (const long long*)d_in[1];   // int64 [2,E]
  const long long* agent = (const long long*)d_in[3];   // int64 [na]
  const float* W0  = (const float*)d_in[4];
  const float* as0 = (const float*)d_in[5];
  const float* ad0 = (const float*)d_in[6];
  const float* b0  = (const float*)d_in[7];
  const float* W1  = (const float*)d_in[8];
  const float* as1 = (const float*)d_in[9];
  const float* ad1 = (const float*)d_in[10];
  const float* b1  = (const float*)d_in[11];

  const int Nn = in_sizes[0] / 256;       // 50000 (multiple of 16)
  const int E  = in_sizes[1] / 2;         // 800000
  const int na = in_sizes[3];             // 16667
  const long long* es = ei;
  const long long* ed = ei + E;
  const int Etot = E + Nn;

  // workspace carve-out
  char* ws = (char*)d_ws;
  size_t off = 0;
  auto carve = [&](size_t bytes) -> void* {
    void* p = ws + off;
    off += (bytes + 255) & ~(size_t)255;
    return p;
  };
  const size_t featBytes = (size_t)Nn * CDIM * sizeof(float);
  float*    XL   = (float*)carve(featBytes);
  float*    H0   = (float*)carve(featBytes);
  float*    H1   = (float*)carve(featBytes);
  float*    Asrc = (float*)carve((size_t)Nn * 2 * 4);
  float*    Adst = (float*)carve((size_t)Nn * 2 * 4);
  unsigned* EMX  = (unsigned*)carve((size_t)Nn * 2 * 4);
  float*    DEN  = (float*)carve((size_t)Nn * 2 * 4);
  float*    EE   = (float*)carve((size_t)Etot * 2 * 4);

  const unsigned NEG_INF_ENC = 0x007FFFFFu;  // enc_f32(-inf)
  dim3 gemmGrid(Nn / 16, CDIM / 32);
  int nodeHB = (2 * Nn + 255) / 256;
  int edgeB  = (Etot + 255) / 256;
  int scatB  = (Etot + 7) / 8;               // 8 waves per 256-thread block
  int featB  = (int)(((size_t)Nn * CDIM + 255) / 256);

  // ---------------- layer 0 ----------------
  gemm_f32_wmma<<<gemmGrid, 64, 0, stream>>>(x, W0, XL, Nn);
  attn_coef<<<nodeHB, 256, 0, stream>>>(XL, as0, ad0, Asrc, Adst, Nn);
  fill_u32<<<nodeHB, 256, 0, stream>>>(EMX, NEG_INF_ENC, (size_t)2 * Nn);
  fill_f32<<<nodeHB, 256, 0, stream>>>(DEN, 0.f, (size_t)2 * Nn);
  fill_f32<<<featB, 256, 0, stream>>>(H0, 0.f, (size_t)Nn * CDIM);
  edge_max_k<<<edgeB, 256, 0, stream>>>(es, ed, Asrc, Adst, EMX, E, Nn);
  edge_exp_k<<<edgeB, 256, 0, stream>>>(es, ed, Asrc, Adst, EMX, EE, DEN, E, Nn);
  edge_scatter_k<<<scatB, 256, 0, stream>>>(es, ed, XL, EE, DEN, H0, E, Nn);
  bias_sigmoid_k<<<featB, 256, 0, stream>>>(H0, b0, (size_t)Nn * CDIM);

  // ---------------- layer 1 ----------------
  gemm_f32_wmma<<<gemmGrid, 64, 0, stream>>>(H0, W1, XL, Nn);
  attn_coef<<<nodeHB, 256, 0, stream>>>(XL, as1, ad1, Asrc, Adst, Nn);
  fill_u32<<<nodeHB, 256, 0, stream>>>(EMX, NEG_INF_ENC, (size_t)2 * Nn);
  fill_f32<<<nodeHB, 256, 0, stream>>>(DEN, 0.f, (size_t)2 * Nn);
  fill_f32<<<featB, 256, 0, stream>>>(H1, 0.f, (size_t)Nn * CDIM);
  edge_max_k<<<edgeB, 256, 0, stream>>>(es, ed, Asrc, Adst, EMX, E, Nn);
  edge_exp_k<<<edgeB, 256, 0, stream>>>(es, ed, Asrc, Adst, EMX, EE, DEN, E, Nn);
  edge_scatter_k<<<scatB, 256, 0, stream>>>(es, ed, XL, EE, DEN, H1, E, Nn);
  bias_sigmoid_k<<<featB, 256, 0, stream>>>(H1, b1, (size_t)Nn * CDIM);

  // ---------------- output gather ----------------
  gather_out_k<<<(na * 512 + 255) / 256, 256, 0, stream>>>(agent, H0, H1, (float*)d_out, na);
}